// VoxelToPixelV2_40295383171317
// MI455X (gfx1250) — compile-verified
//
#include <hip/hip_runtime.h>
#include <hip/hip_bf16.h>

typedef __attribute__((ext_vector_type(2))) float v2f;
typedef __attribute__((ext_vector_type(4))) float v4f;
typedef __attribute__((ext_vector_type(8))) float v8f;
typedef __attribute__((ext_vector_type(4))) int   v4i;
typedef __attribute__((ext_vector_type(4))) unsigned v4u;

#define BIGF 1.0e10f

// ---------------------------------------------------------------------------
// Row r of P = K @ R @ V2P  (3x4), with
//   R   = [[0,-1,0],[0,0,-1],[1,0,0]]
//   V2P = [[.75,0,0,.375],[0,.75,0,-24.625],[0,0,.75,-24.625]]
// (K@R)[r] = [K[r][2], -K[r][0], -K[r][1]]
// ---------------------------------------------------------------------------
__device__ __forceinline__ void p_row(const float* __restrict__ Km, int r, float* o) {
  const float k0 = Km[r * 3 + 0], k1 = Km[r * 3 + 1], k2 = Km[r * 3 + 2];
  const float r0 = k2, r1 = -k0, r2 = -k1;
  o[0] = r0 * 0.75f;
  o[1] = r1 * 0.75f;
  o[2] = r2 * 0.75f;
  o[3] = r0 * 0.375f + r1 * (-24.625f) + r2 * (-24.625f);
}

__device__ __forceinline__ float inv_of(unsigned mdb) {
  const float md = __uint_as_float(mdb);
  return (md < BIGF) ? (1.0f / md) : 0.0f;
}

// ---------------------------------------------------------------------------
// Kernel 1: init per-pixel min-depth (as ordered uint bits) and winner index.
// These arrays are hot (atomics + per-pixel reads) -> regular RT stores.
// ---------------------------------------------------------------------------
__global__ void vtp_init(unsigned* __restrict__ min_bits, int* __restrict__ win_idx,
                         int num_pix, int N) {
  const int i = blockIdx.x * blockDim.x + threadIdx.x;
  if (i < num_pix) {
    min_bits[i] = __float_as_uint(BIGF);
    win_idx[i]  = N;
  }
}

// ---------------------------------------------------------------------------
// Kernel 2: project 16 points per wave via V_WMMA_F32_16X16X4_F32:
//   D(16x16) = A(16x4: rows 0..2 = P, rows 3..15 = 0)
//            x B(4x16: column n = [x_n, y_n, z_n, 1])
// C/D layout => lane l<16 holds p0,p1,p2 of point l in acc[0..2].
// Then per-point z-buffer: atomicMin(min_bits[pix], float_bits(depth)).
// ---------------------------------------------------------------------------
__global__ void vtp_project(const int* __restrict__ coords, const float* __restrict__ Km,
                            const int* __restrict__ pH, const int* __restrict__ pW, int N,
                            unsigned* __restrict__ min_bits, int* __restrict__ point_pix,
                            unsigned* __restrict__ point_dbits) {
  const int lane = threadIdx.x & 31;
  const int wv   = threadIdx.x >> 5;
  const int m    = lane & 15;     // A-row / B-column index
  const int half = lane >> 4;     // 0: K=0,1   1: K=2,3
  const int base = (blockIdx.x * (blockDim.x >> 5) + wv) * 16;
  const int pt   = base + m;
  const int ptc  = (pt < N) ? pt : (N - 1);

  const int4 crd = ((const int4*)coords)[ptc];  // (b, z, y, x)
  const float xf = (float)crd.w, yf = (float)crd.z, zf = (float)crd.y;

  float px, py, pz;
#if __has_builtin(__builtin_amdgcn_wmma_f32_16x16x4_f32)
  // B matrix 4x16 (KxN): lanes 0-15 -> {K0,K1}, lanes 16-31 -> {K2,K3}, N = lane%16
  v2f Bv;
  Bv.x = half ? zf   : xf;
  Bv.y = half ? 1.0f : yf;
  // A matrix 16x4 (MxK): lanes 0-15 -> {K0,K1}, lanes 16-31 -> {K2,K3}, M = lane%16
  float row[4] = {0.0f, 0.0f, 0.0f, 0.0f};
  if (m < 3) p_row(Km, m, row);
  v2f Av;
  Av.x = half ? row[2] : row[0];
  Av.y = half ? row[3] : row[1];

  v8f acc = {};
  acc = __builtin_amdgcn_wmma_f32_16x16x4_f32(
      /*neg_a=*/false, Av, /*neg_b=*/false, Bv,
      /*c_mod=*/(short)0, acc, /*reuse_a=*/false, /*reuse_b=*/false);
  px = acc[0];  // D[0][lane]
  py = acc[1];  // D[1][lane]
  pz = acc[2];  // D[2][lane]  (depth)
#else
  // Exact scalar fallback.
  float r0[4], r1[4], r2[4];
  p_row(Km, 0, r0);
  p_row(Km, 1, r1);
  p_row(Km, 2, r2);
  px = r0[0] * xf + r0[1] * yf + r0[2] * zf + r0[3];
  py = r1[0] * xf + r1[1] * yf + r1[2] * zf + r1[3];
  pz = r2[0] * xf + r2[1] * yf + r2[2] * zf + r2[3];
#endif

  if (lane < 16 && pt < N) {
    const int h = pH[0], w = pW[0];
    const float depth = pz;
    const float safe  = (depth > 1e-6f) ? depth : 1.0f;
    const int u = (int)floorf(px / safe);
    const int v = (int)floorf(py / safe);
    const bool valid = (depth > 1e-6f) && (u >= 0) && (u < w) && (v >= 0) && (v < h);
    if (valid) {
      const int pix = crd.x * (h * w) + v * w + u;
      const unsigned dbits = __float_as_uint(depth);  // positive float: uint order == float order
      atomicMin(&min_bits[pix], dbits);
      point_pix[pt]   = pix;
      point_dbits[pt] = dbits;
    } else {
      point_pix[pt] = -1;  // mark invalid (dbits unread in that case)
    }
  }
}

// ---------------------------------------------------------------------------
// Kernel 3: among depth winners (d <= min_depth[pix]) keep the smallest index.
// ---------------------------------------------------------------------------
__global__ void vtp_select(const int* __restrict__ point_pix,
                           const unsigned* __restrict__ point_dbits,
                           const unsigned* __restrict__ min_bits,
                           int* __restrict__ win_idx, int N) {
  const int i = blockIdx.x * blockDim.x + threadIdx.x;
  if (i >= N) return;
  const int pix = point_pix[i];
  if (pix >= 0 && point_dbits[i] <= min_bits[pix]) {
    atomicMin(&win_idx[pix], i);
  }
}

// ---------------------------------------------------------------------------
// Kernel 4: the ~485MB store-bound phase. Each thread owns 4 consecutive
// w-pixels. Fast path (16B-aligned, within one image row span): gather one
// float4 per winner row per channel-quad, 4x4 register transpose, and emit
// 4 x B128 NON-TEMPORAL stores (512B per wave-store, output is write-once and
// 2.5x the L2 -> keep L2 for win_idx/min_bits/features which have reuse).
// Scalar NT fallback for boundary groups.
// ---------------------------------------------------------------------------
__global__ void vtp_gather(const float* __restrict__ feats, const int* __restrict__ win_idx,
                           const unsigned* __restrict__ min_bits,
                           const int* __restrict__ pH, const int* __restrict__ pW,
                           int N, int C, int num_pix,
                           float* __restrict__ out, float* __restrict__ inv_depth) {
  const int g  = blockIdx.x * blockDim.x + threadIdx.x;
  const int p4 = g * 4;
  if (p4 >= num_pix) return;

  const int HW = pH[0] * pW[0];
  const int C4 = C >> 2;
  const size_t HWs = (size_t)HW;

  const int b0  = p4 / HW;
  const int hw0 = p4 - b0 * HW;

  const bool vec_ok = (p4 + 4 <= num_pix) && ((hw0 & 3) == 0) && (hw0 + 4 <= HW);

  if (vec_ok) {
    const v4i wi = *(const v4i*)(win_idx + p4);
    const v4u mb = *(const v4u*)(min_bits + p4);

    v4f invd;
    invd.x = inv_of(mb.x);
    invd.y = inv_of(mb.y);
    invd.z = inv_of(mb.z);
    invd.w = inv_of(mb.w);
    __builtin_nontemporal_store(invd, (v4f*)(inv_depth + p4));

    const bool h0 = wi.x < N, h1 = wi.y < N, h2 = wi.z < N, h3 = wi.w < N;
    const v4f* r0 = (const v4f*)feats + (size_t)(h0 ? wi.x : 0) * C4;
    const v4f* r1 = (const v4f*)feats + (size_t)(h1 ? wi.y : 0) * C4;
    const v4f* r2 = (const v4f*)feats + (size_t)(h2 ? wi.z : 0) * C4;
    const v4f* r3 = (const v4f*)feats + (size_t)(h3 ? wi.w : 0) * C4;

    const v4f zero = {0.0f, 0.0f, 0.0f, 0.0f};
    size_t o = (size_t)b0 * (size_t)C * HWs + (size_t)hw0;

#pragma unroll 4
    for (int c4 = 0; c4 < C4; ++c4) {
      const v4f f0 = h0 ? r0[c4] : zero;
      const v4f f1 = h1 ? r1[c4] : zero;
      const v4f f2 = h2 ? r2[c4] : zero;
      const v4f f3 = h3 ? r3[c4] : zero;
      // 4x4 transpose: one B128 store per channel, 4 pixels wide
      v4f s0 = {f0.x, f1.x, f2.x, f3.x};
      v4f s1 = {f0.y, f1.y, f2.y, f3.y};
      v4f s2 = {f0.z, f1.z, f2.z, f3.z};
      v4f s3 = {f0.w, f1.w, f2.w, f3.w};
      __builtin_nontemporal_store(s0, (v4f*)(out + o));
      __builtin_nontemporal_store(s1, (v4f*)(out + o + HWs));
      __builtin_nontemporal_store(s2, (v4f*)(out + o + 2 * HWs));
      __builtin_nontemporal_store(s3, (v4f*)(out + o + 3 * HWs));
      o += 4 * HWs;
    }
  } else {
    const int pend = (p4 + 4 < num_pix) ? (p4 + 4) : num_pix;
    for (int p = p4; p < pend; ++p) {
      const int b  = p / HW;
      const int hw = p - b * HW;
      __builtin_nontemporal_store(inv_of(min_bits[p]), inv_depth + p);
      const int win  = win_idx[p];
      const bool has = (win < N);
      const v4f* frow = (const v4f*)feats + (size_t)(has ? win : 0) * C4;
      const v4f zero = {0.0f, 0.0f, 0.0f, 0.0f};
      size_t o = (size_t)b * (size_t)C * HWs + (size_t)hw;
#pragma unroll 4
      for (int c4 = 0; c4 < C4; ++c4) {
        const v4f f = has ? frow[c4] : zero;
        __builtin_nontemporal_store(f.x, out + o);
        __builtin_nontemporal_store(f.y, out + o + HWs);
        __builtin_nontemporal_store(f.z, out + o + 2 * HWs);
        __builtin_nontemporal_store(f.w, out + o + 3 * HWs);
        o += 4 * HWs;
      }
    }
  }
}

// ---------------------------------------------------------------------------
// Launch. Inputs: features(N*C f32), coordinates(N*4 i32), K(9 f32),
// batch_size(1 i32), h(1 i32), w(1 i32). Output: [out (B,C,H,W) | inv_depth
// (B,H,W)] f32 concatenated. Workspace: min_bits | win_idx | point_pix |
// point_dbits (~16.6 MB).
// ---------------------------------------------------------------------------
extern "C" void kernel_launch(void* const* d_in, const int* in_sizes, int n_in,
                              void* d_out, int out_size, void* d_ws, size_t ws_size,
                              hipStream_t stream) {
  const float* features = (const float*)d_in[0];
  const int*   coords   = (const int*)d_in[1];
  const float* Km       = (const float*)d_in[2];
  const int*   pH       = (const int*)d_in[4];
  const int*   pW       = (const int*)d_in[5];
  float* out = (float*)d_out;

  const int N = in_sizes[1] / 4;            // 200000
  const int C = in_sizes[0] / N;            // 64
  const int num_pix = out_size / (C + 1);   // B*H*W

  unsigned* min_bits    = (unsigned*)d_ws;
  int*      win_idx     = (int*)(min_bits + num_pix);
  int*      point_pix   = (int*)(win_idx + num_pix);
  unsigned* point_dbits = (unsigned*)(point_pix + N);

  float* inv_depth = out + (size_t)num_pix * (size_t)C;

  const int T = 256;
  vtp_init<<<(num_pix + T - 1) / T, T, 0, stream>>>(min_bits, win_idx, num_pix, N);

  // 8 waves/block * 16 points/wave = 128 points per block
  const int PTS_PER_BLOCK = (T / 32) * 16;
  vtp_project<<<(N + PTS_PER_BLOCK - 1) / PTS_PER_BLOCK, T, 0, stream>>>(
      coords, Km, pH, pW, N, min_bits, point_pix, point_dbits);

  vtp_select<<<(N + T - 1) / T, T, 0, stream>>>(point_pix, point_dbits, min_bits, win_idx, N);

  const int groups = (num_pix + 3) / 4;  // 4 pixels per thread
  vtp_gather<<<(groups + T - 1) / T, T, 0, stream>>>(
      features, win_idx, min_bits, pH, pW, N, C, num_pix, out, inv_depth);
}